// EfficientMemorySiLU_77489799954632
// MI455X (gfx1250) — compile-verified
//
#include <hip/hip_runtime.h>

// SiLU forward: out = x * sigmoid(x), fp32 -> fp32.
// [4,4096,4096] = 67,108,864 elements. Pure streaming op:
//   537 MB total traffic / 23.3 TB/s  ->  ~23 us roofline floor on MI455X.
// Strategy: 128-bit non-temporal loads/stores (th:TH_LOAD_NT / th:TH_STORE_NT)
// so the one-pass stream does not thrash the 192 MB L2; cheap VALU body using
// v_exp_f32 + v_rcp_f32 (+1 Newton step for accuracy).

typedef float v4f __attribute__((ext_vector_type(4)));

__device__ __forceinline__ float silu_fast(float x) {
    // sigmoid(x) = 1 / (1 + exp(-x)) = rcp(1 + exp2(-x * log2(e)))
    const float LOG2E = 1.4426950408889634f;
    float e = __builtin_amdgcn_exp2f(x * -LOG2E);   // v_exp_f32
    float d = 1.0f + e;
    float r = __builtin_amdgcn_rcpf(d);             // v_rcp_f32 (~1 ulp)
    r = r * (2.0f - d * r);                         // Newton-Raphson refine
    return x * r;
}

__global__ void __launch_bounds__(256)
silu_fwd_nt_kernel(const float* __restrict__ x, float* __restrict__ out, long long n) {
    const long long n4     = n >> 2;                 // # of float4 elements
    const long long gtid   = (long long)blockIdx.x * blockDim.x + threadIdx.x;
    const long long stride = (long long)gridDim.x * blockDim.x;

    const v4f* __restrict__ x4   = (const v4f*)x;
    v4f* __restrict__       out4 = (v4f*)out;

    // Vectorized streaming main loop: global_load_b128 NT -> VALU -> global_store_b128 NT
    for (long long i = gtid; i < n4; i += stride) {
        v4f v = __builtin_nontemporal_load(&x4[i]);
        v4f o;
        o.x = silu_fast(v.x);
        o.y = silu_fast(v.y);
        o.z = silu_fast(v.z);
        o.w = silu_fast(v.w);
        __builtin_nontemporal_store(o, &out4[i]);
    }

    // Scalar tail (n % 4 elements) — empty for 67,108,864 but kept for safety.
    for (long long i = (n4 << 2) + gtid; i < n; i += stride) {
        float v = __builtin_nontemporal_load(&x[i]);
        __builtin_nontemporal_store(silu_fast(v), &out[i]);
    }
}

extern "C" void kernel_launch(void* const* d_in, const int* in_sizes, int n_in,
                              void* d_out, int out_size, void* d_ws, size_t ws_size,
                              hipStream_t stream) {
    const float* x   = (const float*)d_in[0];
    float*       out = (float*)d_out;
    const long long n = (long long)out_size;     // 67,108,864

    const int threads = 256;                     // 8 wave32 waves per block
    const long long n4 = n >> 2;
    // ~4 float4 iterations per thread: enough blocks to flood every WGP while
    // keeping per-thread reuse of the loop body.
    long long blocks = (n4 + (long long)threads * 4 - 1) / ((long long)threads * 4);
    if (blocks < 1) blocks = 1;
    if (blocks > 1u << 20) blocks = 1u << 20;

    silu_fwd_nt_kernel<<<(dim3)(unsigned)blocks, threads, 0, stream>>>(x, out, n);
}